// GraphEncoder_85272280695317
// MI455X (gfx1250) — compile-verified
//
#include <hip/hip_runtime.h>
#include <hip/hip_bf16.h>

typedef __attribute__((ext_vector_type(2))) float v2f;
typedef __attribute__((ext_vector_type(8))) float v8f;

// ---------------------------------------------------------------------------
// Degree / normalization
// ---------------------------------------------------------------------------
__global__ void init_deg_kernel(float* __restrict__ deg, int n) {
  int i = blockIdx.x * blockDim.x + threadIdx.x;
  if (i < n) deg[i] = 1.0f;  // self-loop contributes 1 to every node's degree
}

__global__ void deg_count_kernel(const int* __restrict__ dst,
                                 float* __restrict__ deg, int nE) {
  int e = blockIdx.x * blockDim.x + threadIdx.x;
  if (e < nE) {
    int d = __builtin_nontemporal_load(&dst[e]);
    atomicAdd(&deg[d], 1.0f);
  }
}

__global__ void rsqrt_kernel(float* __restrict__ deg, int n) {
  int i = blockIdx.x * blockDim.x + threadIdx.x;
  if (i < n) {
    float d = deg[i];
    deg[i] = (d > 0.0f) ? rsqrtf(d) : 0.0f;  // deg -> dis in place
  }
}

// ---------------------------------------------------------------------------
// WMMA fp32 GEMM: C[M x NN] = A[M x KK] @ B[KK x NN]   (KK, NN compile-time)
// Optional fused A' = relu(A + bias[k]).
// One wave per 16x16 output tile; K accumulated with V_WMMA_F32_16X16X4_F32.
//
// A 16x4 fp32 frag : lane L holds row M=L&15; VGPR pair = K {kb, kb+1},
//                    kb = k0 + 2*(L>>4)            (ISA 7.12.2, 32-bit A)
// B 4x16 fp32 frag : lane L holds col N=L&15; VGPR pair = K {kb, kb+1}
// C 16x16 fp32     : lane L col N=L&15; VGPR v = row (v + 8*(L>>4))
//
// B is staged in LDS with row stride NN+8 so the two half-wave K-rows hit
// disjoint bank windows (2*LSTR mod 64 == 16): conflict-free ds loads.
// ---------------------------------------------------------------------------
template <int KK, int NN, bool FUSE>
__global__ void wmma_gemm_kernel(const float* __restrict__ A,
                                 const float* __restrict__ B,
                                 float* __restrict__ C,
                                 const float* __restrict__ bias,
                                 int M, int mtiles, int ntiles) {
  constexpr int LSTR = NN + 8;          // bank-conflict padding
  __shared__ float Bs[KK * LSTR];

  // Cooperative stage of B (KK x NN, row-major) into LDS, float4 chunks.
  for (int i = threadIdx.x * 4; i < KK * NN; i += blockDim.x * 4) {
    const int r = i / NN, c = i % NN;   // NN,i multiples of 4: no row cross
    float4 v = *(const float4*)(B + r * NN + c);
    *(float4*)(&Bs[r * LSTR + c]) = v;
  }
  __syncthreads();

  const int wid  = (blockIdx.x * blockDim.x + threadIdx.x) >> 5;  // wave32
  const int lane = threadIdx.x & 31;
  if (wid >= mtiles * ntiles) return;   // wave-uniform: EXEC all-1s at WMMA

  const int mt  = wid / ntiles;
  const int nt  = wid % ntiles;
  const int hh  = lane >> 4;            // 0: K pair {0,1}; 1: K pair {2,3}
  const int l15 = lane & 15;

  int arow = mt * 16 + l15;
  if (arow >= M) arow = M - 1;          // clamp (stores below are guarded)
  const int bcol = nt * 16 + l15;

  v8f acc = {};
#pragma unroll
  for (int k0 = 0; k0 < KK; k0 += 4) {
    const int kb = k0 + (hh << 1);
    float a0 = A[(size_t)arow * KK + kb];
    float a1 = A[(size_t)arow * KK + kb + 1];
    if (FUSE) {
      a0 = fmaxf(a0 + bias[kb], 0.0f);
      a1 = fmaxf(a1 + bias[kb + 1], 0.0f);
    }
    v2f a; a[0] = a0; a[1] = a1;
    v2f b;
    b[0] = Bs[kb * LSTR + bcol];
    b[1] = Bs[(kb + 1) * LSTR + bcol];
    acc = __builtin_amdgcn_wmma_f32_16x16x4_f32(
        /*neg_a=*/false, a, /*neg_b=*/false, b,
        /*c_mod=*/(short)0, acc, /*reuse_a=*/false, /*reuse_b=*/false);
  }

  const int rbase = mt * 16 + (hh << 3);
  if (mt * 16 + 16 <= M) {              // wave-uniform fast path: no guards
    float* Cp = C + (size_t)rbase * NN + bcol;
#pragma unroll
    for (int v = 0; v < 8; ++v) { *Cp = acc[v]; Cp += NN; }
  } else {                              // ragged last tile only
#pragma unroll
    for (int v = 0; v < 8; ++v) {
      const int r = rbase + v;
      if (r < M) C[(size_t)r * NN + bcol] = acc[v];
    }
  }
}

// Scalar fallback GEMM (only used if dims differ from the reference shapes)
template <bool FUSE>
__global__ void naive_gemm_kernel(const float* __restrict__ A,
                                  const float* __restrict__ B,
                                  float* __restrict__ C,
                                  const float* __restrict__ bias,
                                  int M, int K, int Ncols) {
  size_t i = (size_t)blockIdx.x * blockDim.x + threadIdx.x;
  if (i >= (size_t)M * Ncols) return;
  int r = (int)(i / Ncols), c = (int)(i % Ncols);
  float s = 0.0f;
  for (int k = 0; k < K; ++k) {
    float a = A[(size_t)r * K + k];
    if (FUSE) a = fmaxf(a + bias[k], 0.0f);
    s += a * B[(size_t)k * Ncols + c];
  }
  C[i] = s;
}

// ---------------------------------------------------------------------------
// Self-loop message initializes the aggregation buffer:
//   out[i][c] = h[i][c] * dis[i]^2        (plain store, no atomics)
// ---------------------------------------------------------------------------
__global__ void selfloop_init_kernel(const float* __restrict__ h,
                                     const float* __restrict__ dis,
                                     float* __restrict__ out,
                                     int n, int C) {
  size_t i = (size_t)blockIdx.x * blockDim.x + threadIdx.x;
  if (i >= (size_t)n * C) return;
  int node = (int)(i / C);
  float w = dis[node];
  out[i] = h[i] * (w * w);
}

// ---------------------------------------------------------------------------
// Edge scatter: one wave per edge, VEC floats per lane (C == 32*VEC).
// Coalesced 256B/512B row gather of h[src]; scatter-add into out[dst] with
// non-returning global_atomic_add_f32 (resolved in L2; the whole h/out
// working set fits in MI455X's 192MB L2).
// ---------------------------------------------------------------------------
template <int VEC>
__global__ void edge_scatter_kernel(const int* __restrict__ src,
                                    const int* __restrict__ dst,
                                    const float* __restrict__ dis,
                                    const float* __restrict__ h,
                                    float* __restrict__ out,
                                    int nE, int C) {
  int e    = (blockIdx.x * blockDim.x + threadIdx.x) >> 5;
  int lane = threadIdx.x & 31;
  if (e >= nE) return;
  int s = __builtin_nontemporal_load(&src[e]);
  int d = __builtin_nontemporal_load(&dst[e]);
  float w = dis[s] * dis[d];
  const float* hp = h + (size_t)s * C + lane * VEC;
  float*       op = out + (size_t)d * C + lane * VEC;
  if (VEC == 2) {
    float2 v = *(const float2*)hp;
    atomicAdd(&op[0], v.x * w);
    atomicAdd(&op[1], v.y * w);
  } else if (VEC == 4) {
    float4 v = *(const float4*)hp;
    atomicAdd(&op[0], v.x * w);
    atomicAdd(&op[1], v.y * w);
    atomicAdd(&op[2], v.z * w);
    atomicAdd(&op[3], v.w * w);
  } else {
#pragma unroll
    for (int k = 0; k < VEC; ++k) atomicAdd(&op[k], hp[k] * w);
  }
}

// Generic per-channel scatter fallback (non-reference channel counts)
__global__ void edge_scatter_generic_kernel(const int* __restrict__ src,
                                            const int* __restrict__ dst,
                                            const float* __restrict__ dis,
                                            const float* __restrict__ h,
                                            float* __restrict__ out,
                                            int nE, int C) {
  size_t i = (size_t)blockIdx.x * blockDim.x + threadIdx.x;
  if (i >= (size_t)nE * C) return;
  int e = (int)(i / C), c = (int)(i % C);
  int s = src[e], d = dst[e];
  float w = dis[s] * dis[d];
  atomicAdd(&out[(size_t)d * C + c], h[(size_t)s * C + c] * w);
}

// ---------------------------------------------------------------------------
// Final readout: d_out[g][c] = out2[ptr[g]][c] + b2[c]   (bias fused here:
// only the gathered rows ever need it)
// ---------------------------------------------------------------------------
__global__ void gather_out_kernel(const float* __restrict__ agg,
                                  const int* __restrict__ ptr,
                                  const float* __restrict__ b2,
                                  float* __restrict__ out, int C) {
  int g = blockIdx.x;
  int node = ptr[g];
  for (int c = threadIdx.x; c < C; c += blockDim.x)
    out[(size_t)g * C + c] = agg[(size_t)node * C + c] + b2[c];
}

// ---------------------------------------------------------------------------
extern "C" void kernel_launch(void* const* d_in, const int* in_sizes, int n_in,
                              void* d_out, int out_size, void* d_ws, size_t ws_size,
                              hipStream_t stream) {
  const float* x   = (const float*)d_in[0];
  const int*   ei  = (const int*)d_in[1];   // edge_index (2 x nE)
  const int*   ptr = (const int*)d_in[2];   // graph ptr (nG+1)
  const float* W1  = (const float*)d_in[3];
  const float* b1  = (const float*)d_in[4];
  const float* W2  = (const float*)d_in[5];
  const float* b2  = (const float*)d_in[6];

  const int HID = in_sizes[4];              // 64
  const int OUT = in_sizes[6];              // 128
  const int IN  = in_sizes[3] / HID;        // 16
  const int N   = in_sizes[0] / IN;         // 100000
  const int nE  = in_sizes[1] / 2;          // 1600000
  const int nG  = in_sizes[2] - 1;          // 128

  const int* src = ei;
  const int* dst = ei + nE;

  // Workspace layout (floats): dis | h1 | out1 | h2 | out2  (~154 MB)
  float* ws   = (float*)d_ws;
  float* dis  = ws;                  ws += N;
  float* h1   = ws;                  ws += (size_t)N * HID;
  float* out1 = ws;                  ws += (size_t)N * HID;
  float* h2   = ws;                  ws += (size_t)N * OUT;
  float* out2 = ws;

  const int T = 256;

  // --- normalization: deg -> dis = rsqrt(deg) --------------------------------
  init_deg_kernel<<<(N + T - 1) / T, T, 0, stream>>>(dis, N);
  deg_count_kernel<<<(nE + T - 1) / T, T, 0, stream>>>(dst, dis, nE);
  rsqrt_kernel<<<(N + T - 1) / T, T, 0, stream>>>(dis, N);

  const int mtiles = (N + 15) / 16;

  // --- layer 1: h1 = x @ W1 --------------------------------------------------
  if (IN == 16 && HID == 64) {
    int ntl = 64 / 16;
    int blocks = (mtiles * ntl + 3) / 4;     // 4 waves (128 thr) per block
    wmma_gemm_kernel<16, 64, false><<<blocks, 128, 0, stream>>>(
        x, W1, h1, nullptr, N, mtiles, ntl);
  } else {
    naive_gemm_kernel<false><<<((size_t)N * HID + T - 1) / T, T, 0, stream>>>(
        x, W1, h1, nullptr, N, IN, HID);
  }

  // out1 = selfloop + scatter-add over edges
  selfloop_init_kernel<<<((size_t)N * HID + T - 1) / T, T, 0, stream>>>(
      h1, dis, out1, N, HID);
  if (HID == 64) {
    int blocks = (int)(((size_t)nE * 32 + T - 1) / T);
    edge_scatter_kernel<2><<<blocks, T, 0, stream>>>(src, dst, dis, h1, out1, nE, HID);
  } else {
    edge_scatter_generic_kernel<<<((size_t)nE * HID + T - 1) / T, T, 0, stream>>>(
        src, dst, dis, h1, out1, nE, HID);
  }

  // --- layer 2: h2 = relu(out1 + b1) @ W2  (bias+relu fused into A-load) -----
  if (HID == 64 && OUT == 128) {
    int ntl = 128 / 16;
    int blocks = (mtiles * ntl + 3) / 4;
    wmma_gemm_kernel<64, 128, true><<<blocks, 128, 0, stream>>>(
        out1, W2, h2, b1, N, mtiles, ntl);
  } else {
    naive_gemm_kernel<true><<<((size_t)N * OUT + T - 1) / T, T, 0, stream>>>(
        out1, W2, h2, b1, N, HID, OUT);
  }

  selfloop_init_kernel<<<((size_t)N * OUT + T - 1) / T, T, 0, stream>>>(
      h2, dis, out2, N, OUT);
  if (OUT == 128) {
    int blocks = (int)(((size_t)nE * 32 + T - 1) / T);
    edge_scatter_kernel<4><<<blocks, T, 0, stream>>>(src, dst, dis, h2, out2, nE, OUT);
  } else {
    edge_scatter_generic_kernel<<<((size_t)nE * OUT + T - 1) / T, T, 0, stream>>>(
        src, dst, dis, h2, out2, nE, OUT);
  }

  // --- readout: d_out[g] = out2[ptr[g]] + b2 ---------------------------------
  gather_out_kernel<<<nG, 128, 0, stream>>>(out2, ptr, b2, (float*)d_out, OUT);
}